// KDLoss_867583393854
// MI455X (gfx1250) — compile-verified
//
#include <hip/hip_runtime.h>
#include <hip/hip_bf16.h>
#include <math.h>

// ---------------------------------------------------------------------------
// KD distillation loss, MI455X (gfx1250).
//  - One workgroup per (B*S) row; teacher+student rows (2*V fp32 = 250 KB)
//    staged into LDS via the CDNA5 Tensor Data Mover (TENSOR_LOAD_TO_LDS),
//    so HBM is read exactly once while doing an exact 2-pass softmax.
//  - 32 waves each DMA their own 1/32 chunk of both rows, s_wait_tensorcnt,
//    then barrier. Compute passes read LDS as float4 (ds_load_b128).
//  - x = A/Zt - (max_s + log Zs) * (W/Zt); only 2 v_exp_f32 per element.
//  - Deterministic final reduction (no float atomics) in a second kernel.
// ---------------------------------------------------------------------------

typedef unsigned int u32x4 __attribute__((ext_vector_type(4)));
typedef int          i32x8 __attribute__((ext_vector_type(8)));
typedef int          i32x4 __attribute__((ext_vector_type(4)));

#define IGNORE_INDEX (-100)

// Issue one TDM 1-D tile load: nelem fp32 words from gsrc -> LDS byte offset.
// D# layout per CDNA5 ISA ch.8 (group0: ctl/lds/global addr, group1: dims).
// This toolchain's builtin is the 6-arg form:
//   (u32x4 g0, i32x8 g1, i32x4 g2, i32x4 g3, i32x8 extra, i32 cpol)
__device__ __forceinline__ void tdm_load_row_chunk(const float* gsrc,
                                                   unsigned lds_byte_off,
                                                   unsigned nelem) {
  unsigned long long ga = (unsigned long long)(uintptr_t)gsrc;
  u32x4 g0;
  g0[0] = 1u;                                   // count=1 (valid), user mode
  g0[1] = lds_byte_off;                         // lds_addr [63:32]
  g0[2] = (unsigned)(ga & 0xFFFFFFFFull);       // global_addr [95:64]
  g0[3] = (unsigned)((ga >> 32) & 0x1FFFFFFull) // global_addr [120:96]
          | (2u << 30);                         // type=2 ("image") [127:126]
  i32x8 g1;
  g1[0] = (int)(2u << 16);                      // data_size=2 -> 4 bytes
  g1[1] = (int)((nelem & 0xFFFFu) << 16);       // tensor_dim0[15:0]
  g1[2] = (int)(((nelem >> 16) & 0xFFFFu)       // tensor_dim0[31:16]
          | (1u << 16));                        // tensor_dim1 = 1 (low half)
  g1[3] = (int)((nelem & 0xFFFFu) << 16);       // tensor_dim1 hi=0; tile_dim0=nelem
  g1[4] = 1;                                    // tile_dim1=1, tile_dim2=0
  g1[5] = (int)nelem;                           // tensor_dim0_stride[31:0]
  g1[6] = 0;                                    // stride hi / dim1_stride lo
  g1[7] = 0;                                    // dim1_stride hi
  i32x4 z4 = {0, 0, 0, 0};                      // groups 2/3 unused (<=2D)
  i32x8 z8 = {0, 0, 0, 0, 0, 0, 0, 0};
  __builtin_amdgcn_tensor_load_to_lds(g0, g1, z4, z4, z8, 0);
}

template <bool IS_MAX>
__device__ __forceinline__ float block_reduce(float v, float* s_buf) {
  const int lane = threadIdx.x & 31;
  const int wid  = threadIdx.x >> 5;
  const int nw   = (blockDim.x + 31) >> 5;
#pragma unroll
  for (int o = 16; o > 0; o >>= 1) {
    float ov = __shfl_xor(v, o, 32);
    v = IS_MAX ? fmaxf(v, ov) : (v + ov);
  }
  if (lane == 0) s_buf[wid] = v;
  __syncthreads();
  if (wid == 0) {
    float x = (lane < nw) ? s_buf[lane] : (IS_MAX ? -INFINITY : 0.0f);
#pragma unroll
    for (int o = 16; o > 0; o >>= 1) {
      float ov = __shfl_xor(x, o, 32);
      x = IS_MAX ? fmaxf(x, ov) : (x + ov);
    }
    if (lane == 0) s_buf[0] = x;
  }
  __syncthreads();
  float r = s_buf[0];
  __syncthreads();  // s_buf is reused by the next reduction
  return r;
}

__global__ void __launch_bounds__(1024)
kd_row_kernel(const float* __restrict__ logits,          // student [rows*V]
              const float* __restrict__ teacher_logits,  // teacher [rows*V]
              const int*   __restrict__ labels,          // [rows]
              float* __restrict__ ws_x, float* __restrict__ ws_m, int V) {
  extern __shared__ float smem[];      // [0,V): teacher  [V,2V): student
  __shared__ float s_red[32];

  const int row  = blockIdx.x;
  const int tid  = threadIdx.x;
  const int nthr = blockDim.x;

  float* Tt = smem;
  float* Ss = smem + V;

  // ---- Stage both rows into LDS with the TDM, chunked across waves ----
  {
    const int wid    = tid >> 5;
    const int nwaves = nthr >> 5;
    const int chunk  = V / nwaves;
    const int base   = wid * chunk;
    const int len    = (wid == nwaves - 1) ? (V - base) : chunk;
    const unsigned lds0 = (unsigned)(uintptr_t)(void*)smem;  // LDS byte offset
    const size_t   roff = (size_t)row * (size_t)V;
    tdm_load_row_chunk(teacher_logits + roff + base,
                       lds0 + (unsigned)base * 4u, (unsigned)len);
    tdm_load_row_chunk(logits + roff + base,
                       lds0 + (unsigned)(V + base) * 4u, (unsigned)len);
    __builtin_amdgcn_s_wait_tensorcnt(0);  // own chunks landed
  }
  __syncthreads();                         // whole row landed

  // Vectorized view of the LDS rows (V%4 tail handled with scalar loops;
  // when V%4==0 the student row is also 16B-aligned).
  const int nvec = V >> 2;
  const float4* Tv = (const float4*)Tt;
  const float4* Sv = (const float4*)Ss;

  // ---- Pass 1: row maxima for both tensors (ds_load_b128) ----
  float mt = -INFINITY, ms = -INFINITY;
  for (int i = tid; i < nvec; i += nthr) {
    const float4 t4 = Tv[i];
    const float4 s4 = Sv[i];
    mt = fmaxf(mt, fmaxf(fmaxf(t4.x, t4.y), fmaxf(t4.z, t4.w)));
    ms = fmaxf(ms, fmaxf(fmaxf(s4.x, s4.y), fmaxf(s4.z, s4.w)));
  }
  for (int i = (nvec << 2) + tid; i < V; i += nthr) {  // tail (V%4 != 0)
    mt = fmaxf(mt, Tt[i]);
    ms = fmaxf(ms, Ss[i]);
  }
  mt = block_reduce<true>(mt, s_red);
  ms = block_reduce<true>(ms, s_red);

  // ---- Pass 2: Zt, Zs, A = sum_fin e_t*s, W = sum_fin e_t ----
  float zt = 0.0f, zs = 0.0f, acc = 0.0f, accw = 0.0f;
  for (int i = tid; i < nvec; i += nthr) {
    const float4 t4 = Tv[i];
    const float4 s4 = Sv[i];
    const float t[4] = {t4.x, t4.y, t4.z, t4.w};
    const float s[4] = {s4.x, s4.y, s4.z, s4.w};
#pragma unroll
    for (int k = 0; k < 4; ++k) {
      const float et = __expf(t[k] - mt);
      zt += et;
      zs += __expf(s[k] - ms);
      if (!isinf(s[k])) {  // reference masks where student logits are +-inf
        acc  += et * s[k];
        accw += et;
      }
    }
  }
  for (int i = (nvec << 2) + tid; i < V; i += nthr) {  // tail
    const float t = Tt[i];
    const float s = Ss[i];
    const float et = __expf(t - mt);
    zt += et;
    zs += __expf(s - ms);
    if (!isinf(s)) {
      acc  += et * s;
      accw += et;
    }
  }
  zt   = block_reduce<false>(zt, s_red);
  zs   = block_reduce<false>(zs, s_red);
  acc  = block_reduce<false>(acc, s_red);
  accw = block_reduce<false>(accw, s_red);

  if (tid == 0) {
    const float lse = ms + logf(zs);                 // student log-partition
    const float x   = acc / zt - lse * (accw / zt);  // sum p_t * logp_s
    const int lab   = labels[row];
    if (lab != IGNORE_INDEX) {
      ws_x[row] = x;
      ws_m[row] = 1.0f;
    } else {
      ws_x[row] = 0.0f;
      ws_m[row] = 0.0f;
    }
  }
}

__global__ void __launch_bounds__(1024)
kd_finalize(const float* __restrict__ ws_x, const float* __restrict__ ws_m,
            float* __restrict__ out, int rows) {
  __shared__ float s_red[32];
  float sx = 0.0f, sm = 0.0f;
  for (int i = threadIdx.x; i < rows; i += blockDim.x) {
    sx += ws_x[i];
    sm += ws_m[i];
  }
  sx = block_reduce<false>(sx, s_red);
  sm = block_reduce<false>(sm, s_red);
  if (threadIdx.x == 0) out[0] = -sx / sm;
}

extern "C" void kernel_launch(void* const* d_in, const int* in_sizes, int n_in,
                              void* d_out, int out_size, void* d_ws,
                              size_t ws_size, hipStream_t stream) {
  const float* logits         = (const float*)d_in[0];  // [B,S,V] fp32
  const float* teacher_logits = (const float*)d_in[1];  // [B,S,V] fp32
  const int*   labels         = (const int*)d_in[2];    // [B,S] int32

  const int rows = in_sizes[2];                 // B*S
  const int V    = (int)((long long)in_sizes[0] / rows);

  float* ws_x = (float*)d_ws;                   // [rows]
  float* ws_m = ws_x + rows;                    // [rows]

  const size_t shmem = (size_t)2 * (size_t)V * sizeof(float);  // 250 KB @ V=32000

  kd_row_kernel<<<rows, 1024, shmem, stream>>>(logits, teacher_logits, labels,
                                               ws_x, ws_m, V);
  kd_finalize<<<1, 1024, 0, stream>>>(ws_x, ws_m, (float*)d_out, rows);
}